// FWHT_18803366822193
// MI455X (gfx1250) — compile-verified
//
#include <hip/hip_runtime.h>

// FWHT of length 8192 per row = (H16 ⊗ H16 ⊗ H32) x, natural (symmetric) ordering.
// One workgroup (256 threads, 8 waves) per row; fp32 WMMA for the two radix-16
// stages; register-resident radix-32 butterflies for the last 5 stages.

typedef __attribute__((ext_vector_type(2))) float v2f;
typedef __attribute__((ext_vector_type(8))) float v8f;

#define N_COL   8192
#define THREADS 256

__device__ __forceinline__ float hsign(int a, int b) {
    // H[a][b] = (-1)^{popcount(a & b)}
    return (__popc(a & b) & 1) ? -1.0f : 1.0f;
}

__global__ __launch_bounds__(THREADS)
void fwht8192_kernel(const float* __restrict__ x, float* __restrict__ out) {
    __shared__ float lds[N_COL];

    const int t   = threadIdx.x;
    const int row = blockIdx.x;
    const size_t rowoff = (size_t)row * N_COL;

    // ---- load row (coalesced float4) ----
    {
        const float4* __restrict__ src = (const float4*)(x + rowoff);
        float4* dst = (float4*)lds;
        #pragma unroll
        for (int i = 0; i < N_COL / 4 / THREADS; ++i)
            dst[t + i * THREADS] = src[t + i * THREADS];
    }
    __syncthreads();

    const int wave  = t >> 5;
    const int lane  = t & 31;
    const int n     = lane & 15;        // N column (and A's M row) for this lane
    const int khalf = (lane >> 4) << 1; // K sub-offset per lane half (0 or 2)
    const int mbase = (lane >> 4) << 3; // D: lanes 16-31 hold M = vgpr+8

    // ---- phases 1 & 2: wave-local per 256-element block (j2 group) ----
    #pragma unroll
    for (int it = 0; it < 4; ++it) {
        const int blk = wave + it * 8;      // j2 in 0..31, tile index == wave mod 8
        float* base = lds + blk * 256;      // 16x16 tile, row-major [j1][j0]

        // Phase 1: X1[j1][k0] = sum_j0 X[j1][j0] * H16[j0][k0]
        v8f c = {0.f, 0.f, 0.f, 0.f, 0.f, 0.f, 0.f, 0.f};
        #pragma unroll
        for (int ks = 0; ks < 4; ++ks) {
            const int k = ks * 4 + khalf;
            v2f a, b;
            a.x = base[n * 16 + k];         // A[m=n][k]   (A: M across lanes, K in VGPRs)
            a.y = base[n * 16 + k + 1];
            b.x = hsign(k, n);              // B[k][n] = H16
            b.y = hsign(k + 1, n);
            c = __builtin_amdgcn_wmma_f32_16x16x4_f32(
                    false, a, false, b, (short)0, c, false, false);
        }
        #pragma unroll
        for (int v = 0; v < 8; ++v)
            base[(mbase + v) * 16 + n] = c[v];   // D[m=mbase+v][n]

        // Phase 2: X2[k1][k0] = sum_j1 H16[k1][j1] * X1[j1][k0]
        // (same wave wrote X1 above; LDS is in-order within a wave)
        v8f c2 = {0.f, 0.f, 0.f, 0.f, 0.f, 0.f, 0.f, 0.f};
        #pragma unroll
        for (int ks = 0; ks < 4; ++ks) {
            const int k = ks * 4 + khalf;
            v2f a, b;
            a.x = hsign(n, k);              // A[m=n][k] = H16
            a.y = hsign(n, k + 1);
            b.x = base[k * 16 + n];         // B[k][n] = X1[j1=k][k0=n]
            b.y = base[(k + 1) * 16 + n];
            c2 = __builtin_amdgcn_wmma_f32_16x16x4_f32(
                    false, a, false, b, (short)0, c2, false, false);
        }
        #pragma unroll
        for (int v = 0; v < 8; ++v)
            base[(mbase + v) * 16 + n] = c2[v];
    }
    __syncthreads();

    // ---- phase 3: radix-32 FWHT across j2, fully in registers ----
    float vals[32];
    #pragma unroll
    for (int j = 0; j < 32; ++j)
        vals[j] = lds[j * 256 + t];         // conflict-free: 32 consecutive lanes

    #pragma unroll
    for (int s = 1; s < 32; s <<= 1) {
        #pragma unroll
        for (int i = 0; i < 32; ++i) {
            if (!(i & s)) {
                const float a = vals[i];
                const float b = vals[i | s];
                vals[i]     = a + b;
                vals[i | s] = a - b;
            }
        }
    }

    float* __restrict__ o = out + rowoff + t;
    #pragma unroll
    for (int k2 = 0; k2 < 32; ++k2)
        o[k2 * 256] = vals[k2];             // coalesced per k2
}

extern "C" void kernel_launch(void* const* d_in, const int* in_sizes, int n_in,
                              void* d_out, int out_size, void* d_ws, size_t ws_size,
                              hipStream_t stream) {
    (void)n_in; (void)d_ws; (void)ws_size; (void)out_size;
    const float* x = (const float*)d_in[0];
    float* out = (float*)d_out;
    const int rows = in_sizes[0] / N_COL;   // 4096
    fwht8192_kernel<<<rows, THREADS, 0, stream>>>(x, out);
}